// V1Column_34170759807369
// MI455X (gfx1250) — compile-verified
//
#include <hip/hip_runtime.h>
#include <stdint.h>

// Spiking GLIF network forward pass.
// Dims (reference): N=50000, E=2,000,000, T=40, B=1, D=5, N_BASIS=5, n5=5N.
#define NB   5
#define MAXD 5
#define BLK  256

typedef uint32_t v4u __attribute__((ext_vector_type(4)));
typedef uint32_t v8u __attribute__((ext_vector_type(8)));

// ---------------------------------------------------------------------------
// Init: zero circular spike buffer, i_rec planes, psc planes, asc, r; v=v_reset
// ---------------------------------------------------------------------------
__global__ __launch_bounds__(BLK) void init_state_kernel(
    float* __restrict__ zbuf, float* __restrict__ irec, float* __restrict__ psc,
    float* __restrict__ asc,  float* __restrict__ v,    float* __restrict__ r,
    const float* __restrict__ v_reset, int N) {
  int j = blockIdx.x * BLK + threadIdx.x;
  int n5 = NB * N;
  if (j < n5) { zbuf[j] = 0.f; irec[j] = 0.f; psc[j] = 0.f; }
  if (j < 2 * N) asc[j] = 0.f;
  if (j < N) { v[j] = v_reset[j]; r[j] = 0.f; }
}

// ---------------------------------------------------------------------------
// Edge scatter, 4 edges per lane (b128 index loads).
// i_rec is SoA: irec[b*N + post] += z[pre] * weights[e] * edge_basis[e][b].
// zbuf is circular (D,N); slot written at step s is (s mod D); at step t,
// delay d reads slot (t-1-d) mod D (unwritten slots hold the zero init).
// Forward spikes are exactly {0,1}: skip all work when z == 0.
// ---------------------------------------------------------------------------
__device__ __forceinline__ void do_edge(
    int e, int pre, int t, int N,
    const int* __restrict__ post_idx, const float* __restrict__ weights,
    const float* __restrict__ edge_basis, const float* __restrict__ zbuf,
    float* __restrict__ irec) {
  // d = pre / N for D = 5 without integer division
  int d = (pre >= N) + (pre >= 2 * N) + (pre >= 3 * N) + (pre >= 4 * N);
  int n = pre - d * N;
  int slot = (t - 1 - d + MAXD) % MAXD;          // t-1-d >= -5 -> arg >= 0
  float z = zbuf[(size_t)slot * N + n];
  if (z != 0.0f) {
    float w = weights[e] * z;                    // z == 1.0f when spiking
    const float* eb = edge_basis + (size_t)e * NB;
    int post = post_idx[e];
#pragma unroll
    for (int b = 0; b < NB; ++b)
      atomicAdd(&irec[(size_t)b * N + post], w * eb[b]);
  }
}

__global__ __launch_bounds__(BLK) void edge_kernel(
    const int*   __restrict__ pre_idx, const int*   __restrict__ post_idx,
    const float* __restrict__ weights, const float* __restrict__ edge_basis,
    const float* __restrict__ zbuf,    float*       __restrict__ irec,
    int E, int N, int t) {
  int base = (blockIdx.x * BLK + threadIdx.x) * 4;
  if (base >= E) return;
  if (base + 3 < E) {
    int4 p4 = *(const int4*)(pre_idx + base);    // global_load_b128
    do_edge(base + 0, p4.x, t, N, post_idx, weights, edge_basis, zbuf, irec);
    do_edge(base + 1, p4.y, t, N, post_idx, weights, edge_basis, zbuf, irec);
    do_edge(base + 2, p4.z, t, N, post_idx, weights, edge_basis, zbuf, irec);
    do_edge(base + 3, p4.w, t, N, post_idx, weights, edge_basis, zbuf, irec);
  } else {
    for (int e = base; e < E; ++e)
      do_edge(e, pre_idx[e], t, N, post_idx, weights, edge_basis, zbuf, irec);
  }
}

// ---------------------------------------------------------------------------
// Issue a 1D TDM load: 1280 floats  global -> LDS for this block's tile.
// 2-group descriptor; tensor_dim0 = remaining elements so TDM's OOB-zero
// semantics cover the tail block.  Issued by one wave, EXEC ignored.
// ---------------------------------------------------------------------------
__device__ __forceinline__ void tdm_load_row(const float* gsrc, uint32_t lds_addr,
                                             uint32_t remaining) {
  uint64_t ga = (uint64_t)(uintptr_t)gsrc;
  v4u g0;
  g0.x = 1u;                                        // count=1 (valid descriptor)
  g0.y = lds_addr;                                  // LDS byte address
  g0.z = (uint32_t)ga;                              // global_addr[31:0]
  g0.w = ((uint32_t)(ga >> 32) & 0x01ffffffu)       // global_addr[56:32]
         | 0x80000000u;                             // type = 2 ("image")
  v8u g1;
  uint32_t td0 = remaining;                         // tensor_dim0 (OOB -> zero)
  g1[0] = 0x00020000u;                              // data_size=2 (4B), mask=0
  g1[1] = (td0 & 0xffffu) << 16;                    // tensor_dim0[15:0]
  g1[2] = ((td0 >> 16) & 0xffffu)                   // tensor_dim0[31:16]
          | (1u << 16);                             // tensor_dim1 = 1
  g1[3] = (uint32_t)(BLK * NB) << 16;               // tile_dim0 = 1280
  g1[4] = 1u;                                       // tile_dim1=1, tile_dim2=0
  g1[5] = (uint32_t)(BLK * NB);                     // tensor_dim0_stride
  g1[6] = 0u;
  g1[7] = 0u;
  asm volatile("tensor_load_to_lds %0, %1" :: "s"(g0), "s"(g1) : "memory");
}

// ---------------------------------------------------------------------------
// Neuron update.  ext_input tile staged via TDM; syn_decay / psc_initial tiles
// staged via per-lane async global->LDS loads (fixes their stride-20 pattern).
// psc / i_rec / asc are SoA planes for coalesced access; i_rec is reset here
// for the next step's edge scatter.
// ---------------------------------------------------------------------------
__global__ __launch_bounds__(BLK) void neuron_kernel(
    const float* __restrict__ ext,
    const float* __restrict__ decay,      const float* __restrict__ current_factor,
    const float* __restrict__ gathered_g, const float* __restrict__ v_th,
    const float* __restrict__ v_reset,    const float* __restrict__ normalizer,
    const float* __restrict__ t_ref,      const float* __restrict__ exp_dt_k,
    const float* __restrict__ asc_amps,   const float* __restrict__ syn_decay,
    const float* __restrict__ psc_initial,
    float* __restrict__ zbuf, float* __restrict__ irec, float* __restrict__ psc,
    float* __restrict__ asc,  float* __restrict__ v,    float* __restrict__ r,
    float* __restrict__ out, int N, int T, int t) {
  __shared__ float xs[BLK * NB];   // ext_input tile
  __shared__ float sds[BLK * NB];  // syn_decay tile
  __shared__ float pis[BLK * NB];  // psc_initial tile
  const int tid = threadIdx.x;
  const int n5 = NB * N;
  const int rowbase = blockIdx.x * (BLK * NB);
  const float* ext_row = ext + (size_t)t * n5;

  // --- TDM: stage ext_input tile (wave 0 issues; EXEC-independent DMA) ---
  if (tid < 32) {
    uint32_t remaining = (uint32_t)(n5 - rowbase);
    tdm_load_row(ext_row + rowbase, (uint32_t)(uintptr_t)&xs[0], remaining);
  }

  // --- async global->LDS: stage syn_decay / psc_initial tiles ---
#pragma unroll
  for (int k = 0; k < NB; ++k) {
    int idx = tid + k * BLK;                       // 0..1279 within tile
    int g = rowbase + idx;
    if (g >= n5) g = n5 - 1;                       // clamp tail-block OOB
    unsigned lo_sd = (unsigned)(uintptr_t)(&sds[idx]);
    unsigned lo_pi = (unsigned)(uintptr_t)(&pis[idx]);
    const float* gp_sd = syn_decay + g;
    const float* gp_pi = psc_initial + g;
    asm volatile("global_load_async_to_lds_b32 %0, %1, off"
                 :: "v"(lo_sd), "v"(gp_sd) : "memory");
    asm volatile("global_load_async_to_lds_b32 %0, %1, off"
                 :: "v"(lo_pi), "v"(gp_pi) : "memory");
  }

  // Prefetch next timestep's input tile while transfers are in flight.
  if (t + 1 < T) {
    const float* nf = ext + (size_t)(t + 1) * n5 + rowbase + tid * NB;
    __builtin_prefetch(nf, 0, 1);
  }

  asm volatile("s_wait_asynccnt 0x0" ::: "memory");
  if (tid < 32) __builtin_amdgcn_s_wait_tensorcnt(0);
  __syncthreads();

  int n = blockIdx.x * BLK + tid;
  if (n >= N) return;

  // psc update + basis-sum input current; fold in i_rec reset for next step.
  float ic = 0.f;
#pragma unroll
  for (int b = 0; b < NB; ++b) {
    float x   = xs [tid * NB + b];
    float sd  = sds[tid * NB + b];
    float pi  = pis[tid * NB + b];
    size_t j  = (size_t)b * N + n;                 // SoA plane index
    float pj  = psc[j] * sd + (irec[j] + x) * pi;
    psc[j]  = pj;
    irec[j] = 0.f;
    ic += pj;
  }

  int prev_slot = (t - 1 + MAXD) % MAXD;           // z from step t-1
  float pz = zbuf[(size_t)prev_slot * N + n];

  float a0 = exp_dt_k[2 * n + 0] * asc[n]     + pz * asc_amps[2 * n + 0];
  float a1 = exp_dt_k[2 * n + 1] * asc[N + n] + pz * asc_amps[2 * n + 1];
  asc[n]     = a0;
  asc[N + n] = a1;

  float c1 = ic + a0 + a1 + gathered_g[n];
  float nv = decay[n] * v[n] + current_factor[n] * c1;
  if (pz > 0.5f) nv = v_reset[n];

  float vsc = (nv - v_th[n]) / normalizer[n];
  float rr  = r[n];
  float z   = (vsc > 0.0f) ? 1.0f : 0.0f;
  if (rr > 0.0f) z = 0.0f;
  float nr = fmaxf(rr + z * t_ref[n] - 1.0f, 0.0f); // DT = 1.0

  v[n] = nv;
  r[n] = nr;
  zbuf[(size_t)(t % MAXD) * N + n] = z;
  out[(size_t)t * N + n] = z;
}

// ---------------------------------------------------------------------------
// Host launcher
// ---------------------------------------------------------------------------
extern "C" void kernel_launch(void* const* d_in, const int* in_sizes, int n_in,
                              void* d_out, int out_size, void* d_ws, size_t ws_size,
                              hipStream_t stream) {
  const float* weights        = (const float*)d_in[0];
  const float* edge_basis     = (const float*)d_in[1];
  const float* ext_input      = (const float*)d_in[2];
  const float* decay          = (const float*)d_in[3];
  const float* current_factor = (const float*)d_in[4];
  const float* gathered_g     = (const float*)d_in[5];
  const float* v_th           = (const float*)d_in[6];
  const float* v_reset        = (const float*)d_in[7];
  const float* normalizer     = (const float*)d_in[8];
  const float* t_ref          = (const float*)d_in[9];
  const float* exp_dt_k       = (const float*)d_in[10];
  const float* asc_amps       = (const float*)d_in[11];
  const float* syn_decay      = (const float*)d_in[12];
  const float* psc_initial    = (const float*)d_in[13];
  const int*   pre_idx        = (const int*)d_in[14];
  const int*   post_idx       = (const int*)d_in[15];

  const int E  = in_sizes[0];
  const int N  = in_sizes[3];
  const int n5 = in_sizes[12];        // == NB*N == MAXD*N
  const int T  = out_size / N;        // B == 1

  float* out  = (float*)d_out;
  float* ws   = (float*)d_ws;
  float* zbuf = ws;                   // MAXD * N (== n5) floats
  float* irec = zbuf + n5;            // n5 floats, SoA [b][N]
  float* psc  = irec + n5;            // n5 floats, SoA [b][N]
  float* asc  = psc  + n5;            // 2*N floats, SoA [k][N]
  float* vv   = asc  + 2 * N;         // N floats
  float* rr   = vv   + N;             // N floats

  const int init_blocks = (n5 + BLK - 1) / BLK;
  const int edge_groups = (E + 3) / 4;
  const int edge_blocks = (edge_groups + BLK - 1) / BLK;
  const int neur_blocks = (N + BLK - 1) / BLK;

  init_state_kernel<<<init_blocks, BLK, 0, stream>>>(zbuf, irec, psc, asc, vv, rr,
                                                     v_reset, N);
  for (int t = 0; t < T; ++t) {
    edge_kernel<<<edge_blocks, BLK, 0, stream>>>(pre_idx, post_idx, weights,
                                                 edge_basis, zbuf, irec, E, N, t);
    neuron_kernel<<<neur_blocks, BLK, 0, stream>>>(
        ext_input, decay, current_factor, gathered_g, v_th, v_reset, normalizer,
        t_ref, exp_dt_k, asc_amps, syn_decay, psc_initial,
        zbuf, irec, psc, asc, vv, rr, out, N, T, t);
  }
}